// SmplLite_6880537608627
// MI455X (gfx1250) — compile-verified
//
#include <hip/hip_runtime.h>
#include <hip/hip_bf16.h>

// ---------------- problem constants ----------------
#define NJ    24
#define NV    6890
#define NB    10
#define BL    2048            // B*L frames
#define KDIM  224            // 10 betas + 207 pose feats + 1 template + 6 pad
#define KP    (KDIM / 2)     // K-pairs in packed Dirs
#define VPAD  6912           // 432*16
#define NTILE 432
#define FTILE 128

__device__ __constant__ int c_par[NJ] =
  {-1, 0, 0, 0, 1, 2, 3, 4, 5, 6, 7, 8, 9, 9, 9, 12, 13, 14, 16, 17, 18, 19, 20, 21};

typedef __attribute__((ext_vector_type(2))) float v2f;
typedef __attribute__((ext_vector_type(8))) float v8f;

// ---- workspace layout (floats) ----
// Dp (paired-K dirs): 3 * KP * VPAD * 2 = 4,644,864
// F  (frame feats):   BL * KDIM         =   458,752
// A  (frame joints):  BL * NJ * 16      =   786,432  (padded, cols 12..15 = 0)
#define WS_D 0
#define WS_F (3 * KP * VPAD * 2)
#define WS_A (WS_F + BL * KDIM)

// =====================================================================
// Kernel 0: pack paired-K Dirs:  Dp[c][kp][v][h] = Dirs_row(2kp+h)[v]
// =====================================================================
__global__ void smpl_pack_dirs(const float* __restrict__ v_template,
                               const float* __restrict__ shapedirs,
                               const float* __restrict__ posedirs,
                               float* __restrict__ ws) {
  int idx = blockIdx.x * 256 + threadIdx.x;      // 3*KP*VPAD*2 threads
  int h  = idx & 1;
  int t  = idx >> 1;
  int v  = t % VPAD;
  int t2 = t / VPAD;
  int kp = t2 % KP;
  int c  = t2 / KP;
  int k  = 2 * kp + h;
  float val = 0.0f;
  if (v < NV) {
    if (k < NB)        val = shapedirs[v * 30 + c * NB + k];
    else if (k < 217)  val = posedirs[(size_t)(k - NB) * (NV * 3) + v * 3 + c];
    else if (k == 217) val = v_template[v * 3 + c];
  }
  ws[WS_D + idx] = val;
}

// =====================================================================
// Kernel 1: per frame (1 wave): Rodrigues, joint regression, FK chain
//           -> A[frame][24][16] (K-row x padded col) and feature vec F.
// =====================================================================
__global__ void smpl_frame_prep(const float* __restrict__ body_pose,
                                const float* __restrict__ betas,
                                const float* __restrict__ global_orient,
                                const float* __restrict__ J_template,
                                const float* __restrict__ J_shapedirs,
                                float* __restrict__ ws) {
  __shared__ float Jl[NJ][3];
  __shared__ float Tm[NJ][12];
  __shared__ float Ch[NJ][12];

  const int f = blockIdx.x;
  const int j = threadIdx.x;
  float* Fws = ws + WS_F + (size_t)f * KDIM;
  float* Aws = ws + WS_A + (size_t)f * (NJ * 16);

  float R[9];
  if (j < NJ) {
    float a0, a1, a2;
    if (j == 0) {
      a0 = global_orient[f * 3 + 0];
      a1 = global_orient[f * 3 + 1];
      a2 = global_orient[f * 3 + 2];
    } else {
      int o = f * 63 + (j - 1) * 3;
      a0 = body_pose[o + 0]; a1 = body_pose[o + 1]; a2 = body_pose[o + 2];
    }
    float ang = sqrtf(a0 * a0 + a1 * a1 + a2 * a2 + 1e-12f);
    float inv = 1.0f / ang;
    float x = a0 * inv, y = a1 * inv, z = a2 * inv;
    float c = cosf(ang), s = sinf(ang), t = 1.0f - c;
    R[0] = c + t * x * x;      R[1] = -s * z + t * x * y; R[2] = s * y + t * x * z;
    R[3] = s * z + t * x * y;  R[4] = c + t * y * y;      R[5] = -s * x + t * y * z;
    R[6] = -s * y + t * x * z; R[7] = s * x + t * y * z;  R[8] = c + t * z * z;

    float jx = J_template[j * 3 + 0];
    float jy = J_template[j * 3 + 1];
    float jz = J_template[j * 3 + 2];
    for (int k = 0; k < NB; ++k) {
      float b = betas[f * NB + k];
      jx += b * J_shapedirs[j * 30 + 0 * NB + k];
      jy += b * J_shapedirs[j * 30 + 1 * NB + k];
      jz += b * J_shapedirs[j * 30 + 2 * NB + k];
    }
    Jl[j][0] = jx; Jl[j][1] = jy; Jl[j][2] = jz;

    if (j > 0) {
      for (int e = 0; e < 9; ++e)
        Fws[10 + (j - 1) * 9 + e] = R[e] - ((e == 0 || e == 4 || e == 8) ? 1.0f : 0.0f);
    }
  }
  if (j < NB)            Fws[j] = betas[f * NB + j];
  if (j == 24)           Fws[217] = 1.0f;
  if (j >= 25 && j < 31) Fws[218 + (j - 25)] = 0.0f;
  __syncthreads();

  if (j < NJ) {
    int p = (j == 0) ? 0 : c_par[j];
    float rx, ry, rz;
    if (j == 0) { rx = Jl[0][0]; ry = Jl[0][1]; rz = Jl[0][2]; }
    else { rx = Jl[j][0] - Jl[p][0]; ry = Jl[j][1] - Jl[p][1]; rz = Jl[j][2] - Jl[p][2]; }
    Tm[j][0] = R[0]; Tm[j][1] = R[1]; Tm[j][2]  = R[2]; Tm[j][3]  = rx;
    Tm[j][4] = R[3]; Tm[j][5] = R[4]; Tm[j][6]  = R[5]; Tm[j][7]  = ry;
    Tm[j][8] = R[6]; Tm[j][9] = R[7]; Tm[j][10] = R[8]; Tm[j][11] = rz;
  }
  __syncthreads();

  if (j == 0) {
    for (int e = 0; e < 12; ++e) Ch[0][e] = Tm[0][e];
    for (int i = 1; i < NJ; ++i) {
      int p = c_par[i];
      for (int r = 0; r < 3; ++r) {
        float p0 = Ch[p][r * 4 + 0], p1 = Ch[p][r * 4 + 1],
              p2 = Ch[p][r * 4 + 2], p3 = Ch[p][r * 4 + 3];
        for (int cc = 0; cc < 3; ++cc)
          Ch[i][r * 4 + cc] = p0 * Tm[i][0 * 4 + cc] + p1 * Tm[i][1 * 4 + cc] + p2 * Tm[i][2 * 4 + cc];
        Ch[i][r * 4 + 3] = p0 * Tm[i][3] + p1 * Tm[i][7] + p2 * Tm[i][11] + p3;
      }
    }
  }
  __syncthreads();

  if (j < NJ) {
    // row j of B-matrix for LBS WMMA: [R00 R01 R02 t0 | R10.. t1 | R20.. t2 | 0 0 0 0]
    float jx = Jl[j][0], jy = Jl[j][1], jz = Jl[j][2];
    for (int r = 0; r < 3; ++r) {
      float r0 = Ch[j][r * 4 + 0], r1 = Ch[j][r * 4 + 1], r2 = Ch[j][r * 4 + 2];
      float tt = Ch[j][r * 4 + 3] - (r0 * jx + r1 * jy + r2 * jz);
      Aws[j * 16 + r * 4 + 0] = r0;
      Aws[j * 16 + r * 4 + 1] = r1;
      Aws[j * 16 + r * 4 + 2] = r2;
      Aws[j * 16 + r * 4 + 3] = tt;
    }
    Aws[j * 16 + 12] = 0.0f; Aws[j * 16 + 13] = 0.0f;
    Aws[j * 16 + 14] = 0.0f; Aws[j * 16 + 15] = 0.0f;
  }
}

// =====================================================================
// Kernel 2: phase 1: v_posed GEMM (frames x verts, K=224) x3 coords;
//           phase 2: LBS as per-frame WMMA  T_f = W(16x24) @ A_f(24x16)
//           + register-resident combine with (x,y,z,1).
// =====================================================================
__global__ void __launch_bounds__(256)
smpl_lbs_wmma(const float* __restrict__ lbs_w,
              const float* __restrict__ transl,
              const float* __restrict__ ws,
              float* __restrict__ out) {
  __shared__ float Fs[16][KDIM + 1];
  __shared__ __align__(16) float Asp[16 * NJ * 16];   // [frame][j][16]
  __shared__ __align__(16) float Tl[8][16 * 16];      // per-wave T bounce
  __shared__ float Ts[16 * 3];

  const int tid   = threadIdx.x;
  const int wave  = tid >> 5;
  const int lane  = tid & 31;
  const int row   = lane & 15;
  const int half  = (lane < 16) ? 0 : 2;   // K sub-offset for A/B frags
  const int halfp = half >> 1;             // K-pair offset

  const int f0 = blockIdx.y * 16;
  const int n0 = (blockIdx.x * 8 + wave) * 16;
  const int n  = n0 + row;

  const float* Fws = ws + WS_F;
  const float* Aws = ws + WS_A;
  const float2* Dx2 = (const float2*)(ws + WS_D);
  const float2* Dy2 = Dx2 + (size_t)KP * VPAD;
  const float2* Dz2 = Dy2 + (size_t)KP * VPAD;

  // ---- stage F tile, padded A tile, transl ----
  for (int e = tid; e < 16 * KDIM; e += 256) {
    int i = e / KDIM, k = e - i * KDIM;
    Fs[i][k] = Fws[(size_t)(f0 + i) * KDIM + k];
  }
  for (int e = tid; e < 16 * NJ * 16; e += 256)
    Asp[e] = Aws[(size_t)f0 * (NJ * 16) + e];
  if (tid < 48) Ts[tid] = transl[(f0 + tid / 3) * 3 + (tid % 3)];

  // ---- frame-invariant W a-frags for the LBS GEMM (K=24 -> 6 frags) ----
  const int vclamp = (n < NV) ? n : (NV - 1);
  v2f aW[6];
#pragma unroll
  for (int s = 0; s < 6; ++s) {
    aW[s].x = lbs_w[(size_t)vclamp * NJ + 4 * s + half];
    aW[s].y = lbs_w[(size_t)vclamp * NJ + 4 * s + half + 1];
  }
  __syncthreads();

  // ---- phase 1: 56 K-steps x 3 coord planes of V_WMMA_F32_16X16X4_F32 ----
  v8f cx = {}, cy = {}, cz = {};
  for (int k = 0; k < KDIM; k += 4) {
    v2f a;
    a.x = Fs[row][k + half];
    a.y = Fs[row][k + half + 1];
    size_t bi = (size_t)((k >> 1) + halfp) * VPAD + n;   // paired-K, one b64 each
    float2 bx = Dx2[bi], by = Dy2[bi], bz = Dz2[bi];
    v2f vbx, vby, vbz;
    vbx.x = bx.x; vbx.y = bx.y;
    vby.x = by.x; vby.y = by.y;
    vbz.x = bz.x; vbz.y = bz.y;
    cx = __builtin_amdgcn_wmma_f32_16x16x4_f32(false, a, false, vbx, (short)0, cx, false, false);
    cy = __builtin_amdgcn_wmma_f32_16x16x4_f32(false, a, false, vby, (short)0, cy, false, false);
    cz = __builtin_amdgcn_wmma_f32_16x16x4_f32(false, a, false, vbz, (short)0, cz, false, false);
  }

  // ---- phase 2: per-frame LBS GEMM + combine ----
  const int vglob_ok = (n < NV);
  float* Tw = &Tl[wave][0];
#pragma unroll 1
  for (int f = 0; f < 16; ++f) {
    v8f T = {};
#pragma unroll
    for (int s = 0; s < 6; ++s) {
      v2f b;
      b.x = Asp[f * (NJ * 16) + (4 * s + half) * 16 + row];
      b.y = Asp[f * (NJ * 16) + (4 * s + half + 1) * 16 + row];
      T = __builtin_amdgcn_wmma_f32_16x16x4_f32(false, aW[s], false, b, (short)0, T, false, false);
    }
    // scatter T (rows = verts, cols = c*4+d) to per-wave LDS
#pragma unroll
    for (int r = 0; r < 8; ++r)
      Tw[(r + ((lane < 16) ? 0 : 8)) * 16 + row] = T[r];
    asm volatile("s_wait_dscnt 0x0" ::: "memory");

    // combine: half-wave owning frame f in cx/cy/cz registers finishes it
    if ((lane >> 4) == (f >> 3)) {
      int r = f & 7;
      float x = cx[r], y = cy[r], z = cz[r];
      const float4* Tb = (const float4*)&Tw[row * 16];
      float4 t0 = Tb[0], t1 = Tb[1], t2 = Tb[2];
      float ox = t0.x * x + t0.y * y + t0.z * z + t0.w + Ts[f * 3 + 0];
      float oy = t1.x * x + t1.y * y + t1.z * z + t1.w + Ts[f * 3 + 1];
      float oz = t2.x * x + t2.y * y + t2.z * z + t2.w + Ts[f * 3 + 2];
      if (vglob_ok) {
        size_t base = ((size_t)(f0 + f) * NV + n) * 3;
        out[base + 0] = ox;
        out[base + 1] = oy;
        out[base + 2] = oz;
      }
    }
    asm volatile("s_wait_dscnt 0x0" ::: "memory");   // T reads done before next frame overwrites
  }
}

// =====================================================================
extern "C" void kernel_launch(void* const* d_in, const int* in_sizes, int n_in,
                              void* d_out, int out_size, void* d_ws, size_t ws_size,
                              hipStream_t stream) {
  const float* body_pose     = (const float*)d_in[0];
  const float* betas         = (const float*)d_in[1];
  const float* global_orient = (const float*)d_in[2];
  const float* transl        = (const float*)d_in[3];
  const float* v_template    = (const float*)d_in[4];
  const float* shapedirs     = (const float*)d_in[5];
  const float* posedirs      = (const float*)d_in[6];
  const float* lbs_weights   = (const float*)d_in[7];
  const float* J_template    = (const float*)d_in[8];
  const float* J_shapedirs   = (const float*)d_in[9];
  float* out = (float*)d_out;
  float* ws  = (float*)d_ws;

  smpl_pack_dirs<<<(3 * KP * VPAD * 2) / 256, 256, 0, stream>>>(v_template, shapedirs, posedirs, ws);
  smpl_frame_prep<<<BL, 32, 0, stream>>>(body_pose, betas, global_orient,
                                         J_template, J_shapedirs, ws);
  dim3 grid(NTILE / 8, FTILE);
  smpl_lbs_wmma<<<grid, 256, 0, stream>>>(lbs_weights, transl, ws, out);
}